// GaussianSoftmax_67714454389333
// MI455X (gfx1250) — compile-verified
//
#include <hip/hip_runtime.h>

typedef __attribute__((ext_vector_type(2))) float v2f;
typedef __attribute__((ext_vector_type(8))) float v8f;

namespace {
constexpr int Npts    = 4096;   // points per batch
constexpr int Feat    = 16;     // feature dim (= WMMA K total)
constexpr int ROWS    = 16;     // rows per workgroup (one WMMA M tile)
constexpr int STRIDE  = 4100;   // padded LDS row stride (floats): 8*4100 % 64 == 32 -> halves hit disjoint banks
constexpr int THREADS = 512;    // 16 wave32
constexpr int WAVES   = THREADS / 32;
constexpr int CTILES  = Npts / 16;          // 256 column tiles
constexpr int TPW     = CTILES / WAVES;     // 16 tiles per wave
}

extern "C" __global__ __launch_bounds__(THREADS)
void GaussianSoftmax_67714454389333_kernel(const float* __restrict__ X,
                                           const float* __restrict__ sigma,
                                           float* __restrict__ out) {
    extern __shared__ float lds[];
    float* lds_e   = lds;                       // ROWS * STRIDE floats of exp(gaussian)
    float* lds_sum = lds + ROWS * STRIDE;       // 16 row sums
    float* lds_inv = lds_sum + 16;              // 16 reciprocals
    float* lds_sqr = lds_inv + 16;              // 16 row ||x||^2

    const int b       = blockIdx.y;
    const int rowbase = blockIdx.x * ROWS;
    const int tid     = (int)threadIdx.x;
    const int lane    = tid & 31;
    const int wave    = tid >> 5;
    const int h       = lane >> 4;    // which 16-lane half
    const int nl      = lane & 15;    // row/col index within tile

    const float* Xb = X + (size_t)b * Npts * Feat;
    const float inv_sigma = 1.0f / sigma[0];

    // Row ||x||^2 and sum init (threads 0..15)
    if (tid < 16) {
        lds_sum[tid] = 0.0f;
        const float* xr = Xb + (size_t)(rowbase + tid) * Feat;
        float s = 0.0f;
        #pragma unroll
        for (int k = 0; k < Feat; ++k) { float v = xr[k]; s += v * v; }
        lds_sqr[tid] = s;
    }
    __syncthreads();

    // A tile (rows of this slab) in WMMA f32 16x4 layout: 4 K-chunks of 2 VGPRs.
    // Lane (nl + 16h), chunk c holds A[m=nl, k=4c+2h] and A[m=nl, k=4c+2h+1].
    v2f a[4];
    {
        const float* xr = Xb + (size_t)(rowbase + nl) * Feat;
        #pragma unroll
        for (int c = 0; c < 4; ++c) {
            const int k = 4 * c + 2 * h;
            a[c].x = xr[k];
            a[c].y = xr[k + 1];
        }
    }

    // Broadcast row sq for the 8 accumulator rows this lane owns (M = v + 8h)
    float sqr[8];
    #pragma unroll
    for (int v = 0; v < 8; ++v) sqr[v] = lds_sqr[v + 8 * h];

    float psum[8];
    #pragma unroll
    for (int v = 0; v < 8; ++v) psum[v] = 0.0f;

    for (int t = 0; t < TPW; ++t) {
        const int ct = wave + t * WAVES;   // column tile, strided across waves
        const int cb = ct * 16;

        // B tile: B[k, n] = X[cb + n, k]; same per-lane layout as A.
        v2f bm[4];
        const float* xc = Xb + (size_t)(cb + nl) * Feat;
        float sp = 0.0f;
        #pragma unroll
        for (int c = 0; c < 4; ++c) {
            const int k = 4 * c + 2 * h;
            bm[c].x = xc[k];
            bm[c].y = xc[k + 1];
            sp += bm[c].x * bm[c].x + bm[c].y * bm[c].y;
        }
        // lane l and l^16 hold complementary halves of column point (cb+nl)
        const float sqc = sp + __shfl_xor(sp, 16, 32);

        // Gram tile: 4 chained V_WMMA_F32_16X16X4_F32 (K = 16)
        v8f acc = {};
        #pragma unroll
        for (int c = 0; c < 4; ++c)
            acc = __builtin_amdgcn_wmma_f32_16x16x4_f32(
                false, a[c], false, bm[c], (short)0, acc, false, false);

        // Fused distance -> gaussian -> exp; stash in LDS, track row sums
        #pragma unroll
        for (int v = 0; v < 8; ++v) {
            float d = sqr[v] + sqc - 2.0f * acc[v];
            d = fmaxf(d, 0.0f);
            const float g = __expf(-d * inv_sigma);
            const float e = __expf(g);
            lds_e[(v + 8 * h) * STRIDE + cb + nl] = e;
            psum[v] += e;
        }
    }

    // Reduce partial row sums across the 16 lanes of each half, combine via LDS atomics
    #pragma unroll
    for (int v = 0; v < 8; ++v) {
        float r = psum[v];
        r += __shfl_xor(r, 1, 32);
        r += __shfl_xor(r, 2, 32);
        r += __shfl_xor(r, 4, 32);
        r += __shfl_xor(r, 8, 32);
        if (nl == 0) atomicAdd(&lds_sum[v + 8 * h], r);  // ds_add_f32
    }
    __syncthreads();
    if (tid < 16) lds_inv[tid] = 1.0f / lds_sum[tid];
    __syncthreads();

    // Pass 2: normalize and stream out with coalesced b128 stores (only HBM traffic)
    float* outb = out + ((size_t)b * Npts + rowbase) * (size_t)Npts;
    #pragma unroll 1
    for (int r = 0; r < ROWS; ++r) {
        const float inv = lds_inv[r];
        #pragma unroll
        for (int ch = 0; ch < Npts / (THREADS * 4); ++ch) {
            const int col = ch * (THREADS * 4) + tid * 4;
            const float4 ev = *(const float4*)&lds_e[r * STRIDE + col];
            float4 o;
            o.x = ev.x * inv; o.y = ev.y * inv; o.z = ev.z * inv; o.w = ev.w * inv;
            *(float4*)&outb[(size_t)r * Npts + col] = o;
        }
    }
}

extern "C" void kernel_launch(void* const* d_in, const int* in_sizes, int n_in,
                              void* d_out, int out_size, void* d_ws, size_t ws_size,
                              hipStream_t stream) {
    (void)n_in; (void)out_size; (void)d_ws; (void)ws_size;
    const float* X     = (const float*)d_in[0];
    const float* sigma = (const float*)d_in[1];
    float* out         = (float*)d_out;

    const int B = in_sizes[0] / (Npts * Feat);   // 8
    dim3 grid(Npts / ROWS, B);                   // 256 x 8 workgroups
    const size_t shmem = (size_t)(ROWS * STRIDE + 48) * sizeof(float);  // ~256.4 KB (< 320 KB/WGP)

    GaussianSoftmax_67714454389333_kernel<<<grid, dim3(THREADS), shmem, stream>>>(X, sigma, out);
}